// SentenceEncoder_11630771437811
// MI455X (gfx1250) — compile-verified
//
#include <hip/hip_runtime.h>
#include <hip/hip_bf16.h>
#include <math.h>

typedef __attribute__((ext_vector_type(2))) float v2f;
typedef __attribute__((ext_vector_type(8))) float v8f;

#define NN     4096
#define EDIM   1024
#define WFEATC 1024
#define SLOPE  0.01f
#define NEGINF -1e9f

// GEMM blocking: 128M x 64N macro-tile, K-chunk 32, 8 waves (4x2), 32x32 per wave.
#define BM 128
#define BN 64
#define BK 32
#define ASTRIDE 36   // dword row stride for A in LDS (conflict-free b64 fragment reads)
#define PSTR    160  // dword stride per B k-pair row: 64 cols * 2 + 32 pad
                     // (160 mod 64 == 32 -> K+2 half-wave lands in banks 32..63)

// ---------------------------------------------------------------------------
// LDS-tiled fp32 WMMA GEMM core: D[M,N] = A[M,K] @ B[K,N]
// B staged in K-pair-major LDS layout: (k,n) -> Bs[(k>>1)*PSTR + n*2 + (k&1)]
// so each B fragment {B[k][n], B[k+1][n]} is one contiguous b64 read (no movs).
// GATHER=true: A row r of the tile comes from A + rowid_s[r]*lda.
// ---------------------------------------------------------------------------
template <bool GATHER>
__device__ __forceinline__ void gemm_core(
    const float* __restrict__ A, size_t lda, const int* __restrict__ rowid_s,
    const float* __restrict__ B, size_t ldb,
    float* __restrict__ D, size_t ldd,
    int K, int m0, int n0, float* As, float* Bs)
{
    const int tid  = threadIdx.x;
    const int lane = tid & 31;
    const int wave = tid >> 5;
    const int wm   = wave >> 1;          // 0..3
    const int wn   = wave & 1;           // 0..1
    const int half = lane >> 4;          // fragment K-half select
    const int l16  = lane & 15;
    const int wtm  = wm * 32;
    const int wtn  = wn * 32;

    v8f acc00 = {}, acc01 = {}, acc10 = {}, acc11 = {};

    // A global->LDS geometry (coalesced float4)
    const int ar  = tid >> 3;            // 0..31  (row within 32-row pass)
    const int akc = (tid & 7) * 4;       // 0..28  (k-col)

    for (int k0 = 0; k0 < K; k0 += BK) {
        __syncthreads();                 // LDS reuse guard (also covers rowid_s init)
        #pragma unroll
        for (int p = 0; p < 4; ++p) {
            const int r = ar + p * 32;
            const size_t grow = GATHER ? (size_t)rowid_s[r] : (size_t)(m0 + r);
            const float4 v = *reinterpret_cast<const float4*>(A + grow * lda + k0 + akc);
            *reinterpret_cast<float4*>(&As[r * ASTRIDE + akc]) = v;
        }
        // B: 16 k-pairs x 32 col-pairs = 512 units; 2 units per thread.
        #pragma unroll
        for (int p = 0; p < 2; ++p) {
            const int u  = tid + p * 256;
            const int kp = u >> 5;           // 0..15 (k-pair)
            const int cg = u & 31;           // col-pair group
            const float* gb = B + (size_t)(k0 + 2 * kp) * ldb + n0 + cg * 2;
            const float2 lo = *reinterpret_cast<const float2*>(gb);        // row k
            const float2 hi = *reinterpret_cast<const float2*>(gb + ldb);  // row k+1
            float4 w; w.x = lo.x; w.y = hi.x; w.z = lo.y; w.w = hi.y;
            *reinterpret_cast<float4*>(&Bs[kp * PSTR + cg * 4]) = w;
        }
        __syncthreads();

        #pragma unroll
        for (int kk = 0; kk < BK; kk += 4) {
            const int ka = kk + 2 * half;             // even
            const int kp = (kk >> 1) + half;          // B k-pair row
            v2f a0 = *reinterpret_cast<const v2f*>(&As[(wtm + l16) * ASTRIDE + ka]);
            v2f a1 = *reinterpret_cast<const v2f*>(&As[(wtm + 16 + l16) * ASTRIDE + ka]);
            v2f b0 = *reinterpret_cast<const v2f*>(&Bs[kp * PSTR + (wtn + l16) * 2]);
            v2f b1 = *reinterpret_cast<const v2f*>(&Bs[kp * PSTR + (wtn + 16 + l16) * 2]);
            acc00 = __builtin_amdgcn_wmma_f32_16x16x4_f32(false, a0, false, b0, (short)0, acc00, false, false);
            acc01 = __builtin_amdgcn_wmma_f32_16x16x4_f32(false, a0, false, b1, (short)0, acc01, false, false);
            acc10 = __builtin_amdgcn_wmma_f32_16x16x4_f32(false, a1, false, b0, (short)0, acc10, false, false);
            acc11 = __builtin_amdgcn_wmma_f32_16x16x4_f32(false, a1, false, b1, (short)0, acc11, false, false);
        }
    }

    // C/D layout: lanes 0-15 -> rows 0..7, lanes 16-31 -> rows 8..15 of each tile
    float* o = D + (size_t)(m0 + wtm + 8 * half) * ldd + n0 + wtn + l16;
    #pragma unroll
    for (int r = 0; r < 8; ++r) {
        o[(size_t)r * ldd]             = acc00[r];
        o[(size_t)r * ldd + 16]        = acc01[r];
        o[(size_t)(r + 16) * ldd]      = acc10[r];
        o[(size_t)(r + 16) * ldd + 16] = acc11[r];
    }
}

// ---------------------------------------------------------------------------
// K1: h = emb_table[inSen] @ W
// ---------------------------------------------------------------------------
__global__ __launch_bounds__(256) void k_embed_gemm(
    const int*   __restrict__ inSen,
    const float* __restrict__ emb,   // [VOCAB, EDIM]
    const float* __restrict__ W,     // [EDIM, WFEAT]
    float*       __restrict__ h)     // [N, WFEAT]
{
    __shared__ float As[BM * ASTRIDE];
    __shared__ float Bs[(BK / 2) * PSTR];
    __shared__ int   rowid_s[BM];
    const int nb = WFEATC / BN;                    // 16
    const int m0 = (blockIdx.x / nb) * BM;
    const int n0 = (blockIdx.x % nb) * BN;
    if (threadIdx.x < BM) rowid_s[threadIdx.x] = inSen[m0 + threadIdx.x];
    gemm_core<true>(emb, EDIM, rowid_s, W, WFEATC, h, WFEATC, EDIM, m0, n0, As, Bs);
}

// ---------------------------------------------------------------------------
// K2: s_src[i] = h[i,:].a_src ; s_dst[i] = h[i,:].a_dst
// ---------------------------------------------------------------------------
__global__ __launch_bounds__(256) void k_scores(
    const float* __restrict__ h,
    const float* __restrict__ a_src,
    const float* __restrict__ a_dst,
    float*       __restrict__ s_src,
    float*       __restrict__ s_dst)
{
    __shared__ float rs[256], rd[256];
    const int row = blockIdx.x;
    const float* hr = h + (size_t)row * WFEATC;
    float ps = 0.f, pd = 0.f;
    for (int j = threadIdx.x; j < WFEATC; j += 256) {
        const float v = hr[j];
        ps += v * a_src[j];
        pd += v * a_dst[j];
    }
    rs[threadIdx.x] = ps; rd[threadIdx.x] = pd;
    __syncthreads();
    for (int s = 128; s > 0; s >>= 1) {
        if (threadIdx.x < s) {
            rs[threadIdx.x] += rs[threadIdx.x + s];
            rd[threadIdx.x] += rd[threadIdx.x + s];
        }
        __syncthreads();
    }
    if (threadIdx.x == 0) { s_src[row] = rs[0]; s_dst[row] = rd[0]; }
}

// ---------------------------------------------------------------------------
// K3: attention[i,:] = softmax_j( mask(adj, leakyrelu(s_src[i]+s_dst[j])) )
// ---------------------------------------------------------------------------
__global__ __launch_bounds__(256) void k_attention(
    const int*   __restrict__ adj,
    const float* __restrict__ s_src,
    const float* __restrict__ s_dst,
    float*       __restrict__ att)   // [N, N]
{
    __shared__ float ebuf[NN];       // 16 KB
    __shared__ float red[256];
    const int i   = blockIdx.x;
    const int tid = threadIdx.x;
    const float si = s_src[i];
    const int* arow = adj + (size_t)i * NN;

    float mx = -INFINITY;
    for (int j = tid; j < NN; j += 256) {
        float x = si + s_dst[j];
        x = (x >= 0.f) ? x : SLOPE * x;
        x = (arow[j] > 0) ? x : NEGINF;
        ebuf[j] = x;
        mx = fmaxf(mx, x);
    }
    red[tid] = mx;
    __syncthreads();
    for (int s = 128; s > 0; s >>= 1) {
        if (tid < s) red[tid] = fmaxf(red[tid], red[tid + s]);
        __syncthreads();
    }
    mx = red[0];
    __syncthreads();

    float sum = 0.f;
    for (int j = tid; j < NN; j += 256) {
        const float p = __expf(ebuf[j] - mx);
        ebuf[j] = p;
        sum += p;
    }
    red[tid] = sum;
    __syncthreads();
    for (int s = 128; s > 0; s >>= 1) {
        if (tid < s) red[tid] += red[tid + s];
        __syncthreads();
    }
    const float inv = 1.f / red[0];
    __syncthreads();

    float* orow = att + (size_t)i * NN;
    for (int j = tid; j < NN; j += 256)
        orow[j] = ebuf[j] * inv;
}

// ---------------------------------------------------------------------------
// K4: sentence = attention @ h
// ---------------------------------------------------------------------------
__global__ __launch_bounds__(256) void k_sentence_gemm(
    const float* __restrict__ att,   // [N, N]
    const float* __restrict__ h,     // [N, WFEAT]
    float*       __restrict__ sen)   // [N, WFEAT]
{
    __shared__ float As[BM * ASTRIDE];
    __shared__ float Bs[(BK / 2) * PSTR];
    const int nb = WFEATC / BN;                    // 16
    const int m0 = (blockIdx.x / nb) * BM;
    const int n0 = (blockIdx.x % nb) * BN;
    gemm_core<false>(att, NN, nullptr, h, WFEATC, sen, WFEATC, NN, m0, n0, As, Bs);
}

// ---------------------------------------------------------------------------
// K5a: pool partials (strip reduction over rows, coalesced columns)
// ---------------------------------------------------------------------------
__global__ __launch_bounds__(256) void k_pool_part(
    const float* __restrict__ sen,        // [N, WFEAT]
    float*       __restrict__ part)       // [32, WFEAT]
{
    const int j  = blockIdx.x * 256 + threadIdx.x;
    const int r0 = blockIdx.y * (NN / 32);
    float s = 0.f;
    for (int i = 0; i < NN / 32; ++i)
        s += sen[(size_t)(r0 + i) * WFEATC + j];
    part[(size_t)blockIdx.y * WFEATC + j] = s;
}

__global__ __launch_bounds__(256) void k_pool_final(
    const float* __restrict__ part,
    float*       __restrict__ pool)
{
    const int j = blockIdx.x * 256 + threadIdx.x;
    float s = 0.f;
    #pragma unroll
    for (int st = 0; st < 32; ++st)
        s += part[(size_t)st * WFEATC + j];
    pool[j] = s * (1.0f / NN);
}

// ---------------------------------------------------------------------------
// K6: label = softmax(pool @ clf_w + clf_b)
// ---------------------------------------------------------------------------
__global__ __launch_bounds__(256) void k_label(
    const float* __restrict__ pool,
    const float* __restrict__ clf_w,   // [WFEAT, 2]
    const float* __restrict__ clf_b,   // [2]
    float*       __restrict__ label)
{
    __shared__ float r0[256], r1[256];
    float p0 = 0.f, p1 = 0.f;
    for (int j = threadIdx.x; j < WFEATC; j += 256) {
        const float v = pool[j];
        p0 += v * clf_w[j * 2 + 0];
        p1 += v * clf_w[j * 2 + 1];
    }
    r0[threadIdx.x] = p0; r1[threadIdx.x] = p1;
    __syncthreads();
    for (int s = 128; s > 0; s >>= 1) {
        if (threadIdx.x < s) {
            r0[threadIdx.x] += r0[threadIdx.x + s];
            r1[threadIdx.x] += r1[threadIdx.x + s];
        }
        __syncthreads();
    }
    if (threadIdx.x == 0) {
        const float z0 = r0[0] + clf_b[0];
        const float z1 = r1[0] + clf_b[1];
        const float m  = fmaxf(z0, z1);
        const float e0 = __expf(z0 - m);
        const float e1 = __expf(z1 - m);
        const float inv = 1.f / (e0 + e1);
        label[0] = e0 * inv;
        label[1] = e1 * inv;
    }
}

// ---------------------------------------------------------------------------
extern "C" void kernel_launch(void* const* d_in, const int* in_sizes, int n_in,
                              void* d_out, int out_size, void* d_ws, size_t ws_size,
                              hipStream_t stream) {
    const int*   inSen = (const int*)  d_in[0];
    const int*   adj   = (const int*)  d_in[1];
    const float* emb   = (const float*)d_in[2];
    const float* W     = (const float*)d_in[3];
    const float* a_src = (const float*)d_in[4];
    const float* a_dst = (const float*)d_in[5];
    const float* clf_w = (const float*)d_in[6];
    const float* clf_b = (const float*)d_in[7];

    // Output layout: pool[1024] | attention[N*N] | sentence[N*WFEAT] | label[2]
    float* out_pool = (float*)d_out;
    float* out_att  = out_pool + WFEATC;
    float* out_sen  = out_att + (size_t)NN * NN;
    float* out_lab  = out_sen + (size_t)NN * WFEATC;

    // Workspace: h[N*WFEAT] | s_src[N] | s_dst[N] | pool partials[32*WFEAT]
    float* ws_h    = (float*)d_ws;
    float* ws_ssrc = ws_h + (size_t)NN * WFEATC;
    float* ws_sdst = ws_ssrc + NN;
    float* ws_part = ws_sdst + NN;

    const int gemm_blocks = (NN / BM) * (WFEATC / BN);   // 32 * 16 = 512

    k_embed_gemm<<<gemm_blocks, 256, 0, stream>>>(inSen, emb, W, ws_h);
    k_scores<<<NN, 256, 0, stream>>>(ws_h, a_src, a_dst, ws_ssrc, ws_sdst);
    k_attention<<<NN, 256, 0, stream>>>(adj, ws_ssrc, ws_sdst, out_att);
    k_sentence_gemm<<<gemm_blocks, 256, 0, stream>>>(out_att, ws_h, out_sen);
    k_pool_part<<<dim3(WFEATC / 256, 32), 256, 0, stream>>>(out_sen, ws_part);
    k_pool_final<<<WFEATC / 256, 256, 0, stream>>>(ws_part, out_pool);
    k_label<<<1, 256, 0, stream>>>(out_pool, clf_w, clf_b, out_lab);
}